// VanillaAutoencoder_57604101374762
// MI455X (gfx1250) — compile-verified
//
#include <hip/hip_runtime.h>

typedef __attribute__((ext_vector_type(2))) float v2f;
typedef __attribute__((ext_vector_type(8))) float v8f;

#define HSTRIDE 20    // floats per row of 16x16 relayout tile (bank-conflict free)
#define YSTRIDE 132   // floats per row of 16x128 y-update tile (132 % 64 == 4)
#define WPB 4         // waves per block

static __device__ __forceinline__ v8f wmma4(v2f a, v2f b, v8f c) {
  // D = A(16x4) * B(4x16) + C, f32
  return __builtin_amdgcn_wmma_f32_16x16x4_f32(false, a, false, b, (short)0, c, false, false);
}

static __device__ __forceinline__ v8f splat8(float x) {
  v8f r;
#pragma unroll
  for (int i = 0; i < 8; ++i) r[i] = x;
  return r;
}

static __device__ __forceinline__ float fast_tanhf(float x) {
  // tanh(x) = sign(x) * (1 - 2/(exp2(2*log2e*|x|)+1)); exact at 0 and saturation
  const float ax = __builtin_fabsf(x);
  const float e  = __builtin_amdgcn_exp2f(ax * 2.885390081777926814f);
  const float r  = 1.0f - 2.0f * __builtin_amdgcn_rcpf(e + 1.0f);
  return __builtin_copysignf(r, x);
}

static __device__ __forceinline__ v8f tanh8(v8f x) {
  v8f r;
#pragma unroll
  for (int i = 0; i < 8; ++i) r[i] = fast_tanhf(x[i]);
  return r;
}

// Scatter a C-layout 16x16 f32 accumulator to row-major LDS [16][HSTRIDE].
static __device__ __forceinline__ void c_to_lds(float* buf, v8f v, int n16, int hi) {
#pragma unroll
  for (int g = 0; g < 8; ++g) buf[(g + 8 * hi) * HSTRIDE + n16] = v[g];
}

// u = h @ M^T + c  : relayout h (C-layout) -> A-layout via LDS, then 4 WMMAs.
static __device__ __forceinline__ v8f u_from_h(v8f hS, const v2f* MB, float cval,
                                               float* buf, int n16, int hi) {
  c_to_lds(buf, hS, n16, hi);
  asm volatile("s_wait_dscnt 0" ::: "memory");
  v8f acc = splat8(cval);
#pragma unroll
  for (int c = 0; c < 4; ++c) {
    v2f a = *(const v2f*)&buf[n16 * HSTRIDE + 4 * c + 2 * hi];
    acc = wmma4(a, MB[c], acc);
  }
  return acc;
}

// Precompute M = W1 @ W2  (16x16) and c = W1 @ b2 (16) into workspace.
__global__ void dopri5_prep_kernel(const float* __restrict__ W1,
                                   const float* __restrict__ W2,
                                   const float* __restrict__ b2,
                                   float* __restrict__ Mc) {
  const int tid = threadIdx.x;           // 256 threads
  const int i = tid >> 4, k = tid & 15;
  float s = 0.f;
  for (int d = 0; d < 128; ++d) s += W1[i * 128 + d] * W2[d * 16 + k];
  Mc[i * 16 + k] = s;
  if (k == 0) {
    float c = 0.f;
    for (int d = 0; d < 128; ++d) c += W1[i * 128 + d] * b2[d];
    Mc[256 + i] = c;
  }
}

__global__ __launch_bounds__(128)
void dopri5_node_kernel(const float* __restrict__ y0,   // [8192][128]
                        const float* __restrict__ W1,   // [16][128]
                        const float* __restrict__ b1,   // [16]
                        const float* __restrict__ W2,   // [128][16]
                        const float* __restrict__ b2,   // [128]
                        const float* __restrict__ Mc,   // [16][16] M, then [16] c
                        float* __restrict__ out) {      // [50][8192][64]
  __shared__ float ldsY[WPB][16 * YSTRIDE];
  __shared__ float ldsH[WPB][16 * HSTRIDE];

  const int lane = threadIdx.x & 31;
  const int wv   = threadIdx.x >> 5;
  const int tile = blockIdx.x * WPB + wv;   // 512 tiles of 16 batch rows
  const int n16  = lane & 15;
  const int hi   = lane >> 4;
  float* bufY = ldsY[wv];
  float* bufH = ldsH[wv];

  // Per-lane broadcast constants (C-layout columns index by n16)
  const float b1s = b1[n16];
  const float cs  = Mc[256 + n16];
  float b2s[8];
#pragma unroll
  for (int t = 0; t < 8; ++t) b2s[t] = b2[16 * t + n16];

  // B-layout operands: lane holds N=n16, K(in-chunk) = 2*hi + v
  v2f MB[4];
#pragma unroll
  for (int c = 0; c < 4; ++c) MB[c] = *(const v2f*)&Mc[n16 * 16 + 4 * c + 2 * hi];

  v2f W1B[32];  // B = W1^T chunk: W1[n][4c+2hi+v]
#pragma unroll
  for (int c = 0; c < 32; ++c) W1B[c] = *(const v2f*)&W1[n16 * 128 + 4 * c + 2 * hi];

  v2f W2B[8][4];  // B = W2^T tile t chunk c: W2[16t+n][4c+2hi+v]
#pragma unroll
  for (int t = 0; t < 8; ++t)
#pragma unroll
    for (int c = 0; c < 4; ++c)
      W2B[t][c] = *(const v2f*)&W2[(16 * t + n16) * 16 + 4 * c + 2 * hi];

  // State tile in A-layout: lane holds M=n16, K = 4c + 2hi + v
  v2f yA[32];
  {
    const float* yb = y0 + (size_t)(tile * 16 + n16) * 128 + 2 * hi;
#pragma unroll
    for (int c = 0; c < 32; ++c) yA[c] = *(const v2f*)(yb + 4 * c);
  }

  const size_t outStride = (size_t)8192 * 64;

  // t = 0 output: first 64 state dims (chunks 0..15)
  {
    float* o = out + ((size_t)tile * 16 + n16) * 64 + 2 * hi;
#pragma unroll
    for (int c = 0; c < 16; ++c) *(v2f*)(o + 4 * c) = yA[c];
  }

  for (int iv = 0; iv < 49; ++iv) {
    const float t0 = (float)iv * (1.0f / 49.0f);
    const float t1 = (float)(iv + 1) * (1.0f / 49.0f);
    const float hstep = (t1 - t0) * 0.5f;

    for (int sub = 0; sub < 2; ++sub) {
      // ---- z0 = y @ W1^T + b1 : 32 accumulating WMMAs, 4 chains for ILP
      v8f a0 = splat8(b1s), a1 = splat8(0.f), a2 = splat8(0.f), a3 = splat8(0.f);
#pragma unroll
      for (int c = 0; c < 32; c += 4) {
        a0 = wmma4(yA[c + 0], W1B[c + 0], a0);
        a1 = wmma4(yA[c + 1], W1B[c + 1], a1);
        a2 = wmma4(yA[c + 2], W1B[c + 2], a2);
        a3 = wmma4(yA[c + 3], W1B[c + 3], a3);
      }
      const v8f z0 = (a0 + a1) + (a2 + a3);

      // ---- 6 dopri5 stages, entirely in 16-dim space
      v8f hS = tanh8(z0);
      v8f H  = hS * (float)(35.0 / 384.0);
      const v8f u0 = u_from_h(hS, MB, cs, bufH, n16, hi);

      v8f z = z0 + (hstep * (float)(1.0 / 5.0)) * u0;
      hS = tanh8(z);                                     // b_1 == 0
      const v8f u1 = u_from_h(hS, MB, cs, bufH, n16, hi);

      z = z0 + hstep * (u0 * (float)(3.0 / 40.0) + u1 * (float)(9.0 / 40.0));
      hS = tanh8(z);
      H += hS * (float)(500.0 / 1113.0);
      const v8f u2 = u_from_h(hS, MB, cs, bufH, n16, hi);

      z = z0 + hstep * (u0 * (float)(44.0 / 45.0) + u1 * (float)(-56.0 / 15.0) +
                        u2 * (float)(32.0 / 9.0));
      hS = tanh8(z);
      H += hS * (float)(125.0 / 192.0);
      const v8f u3 = u_from_h(hS, MB, cs, bufH, n16, hi);

      z = z0 + hstep * (u0 * (float)(19372.0 / 6561.0) + u1 * (float)(-25360.0 / 2187.0) +
                        u2 * (float)(64448.0 / 6561.0) + u3 * (float)(-212.0 / 729.0));
      hS = tanh8(z);
      H += hS * (float)(-2187.0 / 6784.0);
      const v8f u4 = u_from_h(hS, MB, cs, bufH, n16, hi);

      z = z0 + hstep * (u0 * (float)(9017.0 / 3168.0) + u1 * (float)(-355.0 / 33.0) +
                        u2 * (float)(46732.0 / 5247.0) + u3 * (float)(49.0 / 176.0) +
                        u4 * (float)(-5103.0 / 18656.0));
      hS = tanh8(z);
      H += hS * (float)(11.0 / 84.0);

      // ---- y += hstep * (H @ W2^T + b2)
      c_to_lds(bufH, H, n16, hi);
      asm volatile("s_wait_dscnt 0" ::: "memory");
      v2f HA[4];
#pragma unroll
      for (int c = 0; c < 4; ++c)
        HA[c] = *(const v2f*)&bufH[n16 * HSTRIDE + 4 * c + 2 * hi];

#pragma unroll
      for (int t = 0; t < 8; ++t) {
        v8f D = splat8(b2s[t]);
#pragma unroll
        for (int c = 0; c < 4; ++c) D = wmma4(HA[c], W2B[t][c], D);
#pragma unroll
        for (int g = 0; g < 8; ++g)
          bufY[(g + 8 * hi) * YSTRIDE + 16 * t + n16] = hstep * D[g];
      }
      asm volatile("s_wait_dscnt 0" ::: "memory");
#pragma unroll
      for (int c = 0; c < 32; ++c)
        yA[c] += *(const v2f*)&bufY[n16 * YSTRIDE + 4 * c + 2 * hi];
    }

    // ---- emit trajectory slice for time iv+1 (first 64 dims)
    float* o = out + (size_t)(iv + 1) * outStride + ((size_t)tile * 16 + n16) * 64 + 2 * hi;
#pragma unroll
    for (int c = 0; c < 16; ++c) *(v2f*)(o + 4 * c) = yA[c];
  }
}

extern "C" void kernel_launch(void* const* d_in, const int* in_sizes, int n_in,
                              void* d_out, int out_size, void* d_ws, size_t ws_size,
                              hipStream_t stream) {
  const float* y0 = (const float*)d_in[0];
  const float* W1 = (const float*)d_in[1];
  const float* b1 = (const float*)d_in[2];
  const float* W2 = (const float*)d_in[3];
  const float* b2 = (const float*)d_in[4];
  float* Mc = (float*)d_ws;  // 272 floats: M (256) + c (16)

  dopri5_prep_kernel<<<1, 256, 0, stream>>>(W1, W2, b2, Mc);
  // 8192 batch rows / 16 per wave = 512 waves; 4 waves (128 threads) per block
  dopri5_node_kernel<<<128, 128, 0, stream>>>(y0, W1, b1, W2, b2, Mc, (float*)d_out);
}